// PairTabAtomicModel_25537875542623
// MI455X (gfx1250) — compile-verified
//
#include <hip/hip_runtime.h>
#include <stdint.h>

// Problem geometry (fixed by the reference's setup_inputs()).
#define NF        8
#define NLOC      8192
#define NNEI      128
#define NALL      16384
#define NTYPES    4
#define NSPLINE   2000
#define RCUT_F    6.0f

#define THREADS          1024
#define BLOCKS_PER_FRAME 32
#define ATOMS_PER_BLOCK  (NLOC / BLOCKS_PER_FRAME)   // 256
#define WAVES            (THREADS / 32)              // 32
#define ATOMS_PER_WAVE   (ATOMS_PER_BLOCK / WAVES)   // 8

// ---- CDNA5 async global->LDS copy (ASYNCcnt path) ----------------------
// GVS mode: scalar 64-bit base plus per-lane 32-bit byte offset; each lane copies 16 B into LDS.
__device__ __forceinline__ void async_copy_b128(uint32_t lds_byte_addr,
                                                const void* sgpr_base,
                                                uint32_t byte_off) {
  asm volatile("global_load_async_to_lds_b128 %0, %1, %2 offset:0"
               :
               : "v"(lds_byte_addr), "v"(byte_off), "s"(sgpr_base)
               : "memory");
}

__device__ __forceinline__ void wait_async_zero() {
#if __has_builtin(__builtin_amdgcn_s_wait_asynccnt)
  __builtin_amdgcn_s_wait_asynccnt(0);
#else
  asm volatile("s_wait_asynccnt 0" ::: "memory");
#endif
}

__global__ __launch_bounds__(THREADS, 1)
void PairTabAtomicModel_kernel(const float*  __restrict__ coord,
                               const float4* __restrict__ tab4,
                               const float*  __restrict__ tab_info,
                               const int*    __restrict__ atype,
                               const int4*   __restrict__ nlist4,
                               float*        __restrict__ out) {
  // 192 KB coords + 64 KB types = 256 KB <= 320 KB WGP LDS (CDNA5-only).
  __shared__ __align__(16) float s_coord[NALL * 3];
  __shared__ __align__(16) int   s_atype[NALL];

  const int f   = blockIdx.x / BLOCKS_PER_FRAME;
  const int blk = blockIdx.x % BLOCKS_PER_FRAME;
  const int tid = threadIdx.x;

  const float* gcoord = coord + (size_t)f * (NALL * 3);
  const int*   gat    = atype + (size_t)f * NALL;

  // Stage one frame's coordinates + types into LDS with async b128 copies.
  // Low 32 bits of a flat LDS pointer == LDS byte offset (aperture layout).
  const uint32_t lds_c = (uint32_t)(uintptr_t)(const void*)&s_coord[0];
  const uint32_t lds_a = (uint32_t)(uintptr_t)(const void*)&s_atype[0];

#pragma unroll
  for (int c = 0; c < (NALL * 3 * 4) / 16 / THREADS; ++c) {   // 12 chunks/thread
    const uint32_t off = (uint32_t)(tid + c * THREADS) * 16u;
    async_copy_b128(lds_c + off, (const void*)gcoord, off);
  }
#pragma unroll
  for (int c = 0; c < (NALL * 4) / 16 / THREADS; ++c) {       // 4 chunks/thread
    const uint32_t off = (uint32_t)(tid + c * THREADS) * 16u;
    async_copy_b128(lds_a + off, (const void*)gat, off);
  }
  wait_async_zero();
  __syncthreads();

  const float rmin   = tab_info[0];
  const float hh     = tab_info[1];
  const float inv_hh = 1.0f / hh;
  const float rmax   = rmin + (float)NSPLINE * hh;

  const int wave = tid >> 5;
  const int lane = tid & 31;

  for (int t = 0; t < ATOMS_PER_WAVE; ++t) {
    const int a = blk * ATOMS_PER_BLOCK + wave * ATOMS_PER_WAVE + t;

    const int   itype = s_atype[a];
    const float xl = s_coord[3 * a + 0];
    const float yl = s_coord[3 * a + 1];
    const float zl = s_coord[3 * a + 2];

    // 128 neighbors / wave32: each lane takes 4 contiguous (coalesced int4).
    const int4 jj = nlist4[((size_t)f * NLOC + a) * (NNEI / 4) + lane];
    const int js[4] = {jj.x, jj.y, jj.z, jj.w};

    float sum = 0.0f;
#pragma unroll
    for (int k = 0; k < 4; ++k) {
      const int  j  = js[k];
      const bool m  = (j >= 0);
      const int  mj = m ? j : 0;

      const float dx = s_coord[3 * mj + 0] - xl;
      const float dy = s_coord[3 * mj + 1] - yl;
      const float dz = s_coord[3 * mj + 2] - zl;
      const float rr = sqrtf(fmaf(dx, dx, fmaf(dy, dy, dz * dz)));

      float uu  = m ? (rr - rmin) * inv_hh : (float)(NSPLINE + 1);
      int   idx = (int)uu;                       // uu >= 0 in all live cases
      const float fr = uu - (float)idx;
      int clip = idx < 0 ? 0 : (idx > NSPLINE - 1 ? NSPLINE - 1 : idx);

      const int jtype = s_atype[mj];
      float4 cf = tab4[(size_t)((itype * NTYPES + jtype) * NSPLINE + clip)];
      if (idx > NSPLINE) { cf.x = 0.f; cf.y = 0.f; cf.z = 0.f; cf.w = 0.f; }

      float e = fmaf(fmaf(fmaf(cf.x, fr, cf.y), fr, cf.z), fr, cf.w);
      if (rr >= RCUT_F || rr >= rmax) e = 0.0f;
      if (!m)                          e = 0.0f;
      sum += e;
    }

    // wave32 butterfly reduction (5 steps on CDNA5 vs 6 on wave64).
#pragma unroll
    for (int s = 16; s > 0; s >>= 1) sum += __shfl_xor(sum, s, 32);

    if (lane == 0) out[(size_t)f * NLOC + a] = 0.5f * sum;
  }
}

extern "C" void kernel_launch(void* const* d_in, const int* in_sizes, int n_in,
                              void* d_out, int out_size, void* d_ws, size_t ws_size,
                              hipStream_t stream) {
  (void)in_sizes; (void)n_in; (void)out_size; (void)d_ws; (void)ws_size;
  const float*  coord    = (const float*)d_in[0];
  const float4* tab4     = (const float4*)d_in[1];
  const float*  tab_info = (const float*)d_in[2];
  const int*    atype    = (const int*)d_in[3];
  const int4*   nlist4   = (const int4*)d_in[4];
  float*        out      = (float*)d_out;

  dim3 grid(NF * BLOCKS_PER_FRAME);
  dim3 block(THREADS);
  hipLaunchKernelGGL(PairTabAtomicModel_kernel, grid, block, 0, stream,
                     coord, tab4, tab_info, atype, nlist4, out);
}